// Attention_27977416966176
// MI455X (gfx1250) — compile-verified
//
#include <hip/hip_runtime.h>

// MI455X / gfx1250, wave32. Memory-bound problem (0.5 FLOP/byte); one-pass
// online-softmax structure reads enc (512 MiB) from HBM once (~22 us floor at
// 23.3 TB/s). Scores computed with V_WMMA_F32_16X16X4_F32 (fp32 WMMA, free
// since HBM-bound; N-dimension replication waste costs nothing).

typedef __attribute__((ext_vector_type(2))) float v2f;
typedef __attribute__((ext_vector_type(8))) float v8f;

#define B_ 64
#define S_ 2048
#define H_ 1024
#define L_ 4
#define CHUNK_ 128                 // seq rows per workgroup (8 waves x 16 rows)
#define NCHUNK_ (S_ / CHUNK_)      // 16 chunks per batch

// ---------------------------------------------------------------------------
// K0: havg[b,h] = mean_l dec[l,b,h]   (dec layout [L,B,H])
// ---------------------------------------------------------------------------
__global__ __launch_bounds__(256) void k_havg(const float* __restrict__ dec,
                                              float* __restrict__ havg) {
  int idx = blockIdx.x * 256 + threadIdx.x;  // idx = b*H + h, B_*H_ threads
  float s = 0.0f;
#pragma unroll
  for (int l = 0; l < L_; ++l) s += dec[(size_t)l * (B_ * H_) + idx];
  havg[idx] = s * (1.0f / (float)L_);
}

// ---------------------------------------------------------------------------
// K1: per (b, chunk): WMMA scores -> chunk softmax partial -> weighted acc.
// Emits:  acc_out[bx*H + h] = sum_s exp(score_s - m_chunk) * enc[b,s,h]
//         meta_m[bx] = m_chunk, meta_l[bx] = sum_s exp(score_s - m_chunk)
// ---------------------------------------------------------------------------
__global__ __launch_bounds__(256) void k_scores_partial(
    const float* __restrict__ enc, const float* __restrict__ havg,
    float* __restrict__ acc_out, float* __restrict__ meta_m,
    float* __restrict__ meta_l) {
  __shared__ float lds_havg[H_];      // 4 KB: havg[b,:]
  __shared__ float lds_sc[CHUNK_];    // 512 B: scores, then exp(scores - m)

  const int bx    = blockIdx.x;
  const int b     = bx / NCHUNK_;
  const int chunk = bx % NCHUNK_;
  const int s0    = chunk * CHUNK_;
  const int tid   = threadIdx.x;
  const int wave  = tid >> 5;
  const int lane  = tid & 31;

  // Stage havg[b,:] into LDS (coalesced float4, 256 threads x 16B = 4KB).
  {
    const float4* src = (const float4*)(havg + (size_t)b * H_);
    ((float4*)lds_havg)[tid] = src[tid];
  }
  __syncthreads();

  // ---- Phase 1: scores via chained V_WMMA_F32_16X16X4_F32 ----------------
  // Wave w owns rows s0 + w*16 .. +15.  A layout (16x4 f32): lanes 0-15 hold
  // K=0,1 in v0,v1; lanes 16-31 hold K=2,3  -> per-lane float2 at
  // (row = s0+w*16+(lane&15), h = h0 + (lane>>4)*2).  B (4x16) = havg[h0+k]
  // replicated across all 16 columns: per-lane float2 broadcast from LDS.
  {
    const int halfk = lane >> 4;   // 0 or 1
    const int mrow  = lane & 15;
    const float* Abase =
        enc + ((size_t)b * S_ + (size_t)(s0 + wave * 16 + mrow)) * H_ + halfk * 2;
    v8f acc = {0.f, 0.f, 0.f, 0.f, 0.f, 0.f, 0.f, 0.f};
#pragma unroll 8
    for (int h0 = 0; h0 < H_; h0 += 4) {
      v2f a    = *(const v2f*)(Abase + h0);
      v2f bmat = *(const v2f*)(lds_havg + h0 + halfk * 2);
      acc = __builtin_amdgcn_wmma_f32_16x16x4_f32(
          /*neg_a=*/false, a, /*neg_b=*/false, bmat,
          /*c_mod=*/(short)0, acc, /*reuse_a=*/false, /*reuse_b=*/false);
    }
    // D layout: VGPR j = rows {j | j+8}, N = lane | lane-16.  All columns
    // identical (B replicated) -> lane N==0 of each half extracts 8 rows.
    if (mrow == 0) {
      int rbase = wave * 16 + halfk * 8;
#pragma unroll
      for (int j = 0; j < 8; ++j) lds_sc[rbase + j] = acc[j];
    }
  }
  __syncthreads();

  // ---- Phase 2: chunk max / exp / sum (serial => deterministic) ----------
  float m_loc = -3.402823466e+38f;
  for (int i = 0; i < CHUNK_; ++i) m_loc = fmaxf(m_loc, lds_sc[i]);
  __syncthreads();
  if (tid < CHUNK_) lds_sc[tid] = __expf(lds_sc[tid] - m_loc);
  __syncthreads();
  float l_loc = 0.0f;
  for (int i = 0; i < CHUNK_; ++i) l_loc += lds_sc[i];
  if (tid == 0) { meta_m[bx] = m_loc; meta_l[bx] = l_loc; }

  // ---- Phase 3: acc[h] = sum_s e_s * enc[b,s,h] --------------------------
  // Coalesced: thread owns 4 consecutive h (float4); chunk rows are L2-hot.
  float4 acc4 = {0.f, 0.f, 0.f, 0.f};
  const float4* encrow = (const float4*)(enc + ((size_t)b * S_ + s0) * H_);
  for (int s = 0; s < CHUNK_; ++s) {
    float  e = lds_sc[s];
    float4 v = encrow[(size_t)s * (H_ / 4) + tid];
    acc4.x += e * v.x; acc4.y += e * v.y;
    acc4.z += e * v.z; acc4.w += e * v.w;
  }
  ((float4*)(acc_out + (size_t)bx * H_))[tid] = acc4;
}

// ---------------------------------------------------------------------------
// K2: combine chunk partials per batch with standard (m, l, acc) rescaling.
// out[b,0,h] = (sum_i exp(m_i - m) * acc_i[h]) / (sum_i l_i * exp(m_i - m))
// ---------------------------------------------------------------------------
__global__ __launch_bounds__(256) void k_combine(
    const float* __restrict__ acc_in, const float* __restrict__ meta_m,
    const float* __restrict__ meta_l, float* __restrict__ out) {
  const int b   = blockIdx.x;
  const int tid = threadIdx.x;
  __shared__ float sscale[NCHUNK_];

  float m = -3.402823466e+38f;
  for (int i = 0; i < NCHUNK_; ++i) m = fmaxf(m, meta_m[b * NCHUNK_ + i]);
  float Lsum = 0.0f;
  for (int i = 0; i < NCHUNK_; ++i)
    Lsum += meta_l[b * NCHUNK_ + i] * __expf(meta_m[b * NCHUNK_ + i] - m);
  if (tid < NCHUNK_) sscale[tid] = __expf(meta_m[b * NCHUNK_ + tid] - m);
  __syncthreads();

  float4 acc = {0.f, 0.f, 0.f, 0.f};
  for (int i = 0; i < NCHUNK_; ++i) {
    float sc = sscale[i];
    float4 v = ((const float4*)(acc_in + ((size_t)b * NCHUNK_ + i) * H_))[tid];
    acc.x += sc * v.x; acc.y += sc * v.y;
    acc.z += sc * v.z; acc.w += sc * v.w;
  }
  float inv = 1.0f / Lsum;
  float4 r = {acc.x * inv, acc.y * inv, acc.z * inv, acc.w * inv};
  ((float4*)(out + (size_t)b * H_))[tid] = r;
}

// ---------------------------------------------------------------------------
extern "C" void kernel_launch(void* const* d_in, const int* in_sizes, int n_in,
                              void* d_out, int out_size, void* d_ws,
                              size_t ws_size, hipStream_t stream) {
  const float* enc = (const float*)d_in[0];  // [B,S,H] fp32
  const float* dec = (const float*)d_in[1];  // [L,B,H] fp32
  float* out = (float*)d_out;                // [B,1,H] fp32

  // Workspace layout (~4.46 MB total):
  float* ws     = (float*)d_ws;
  float* acc_p  = ws;                                   // B*NCHUNK*H floats
  float* meta_m = acc_p + (size_t)B_ * NCHUNK_ * H_;    // B*NCHUNK
  float* meta_l = meta_m + (size_t)B_ * NCHUNK_;        // B*NCHUNK
  float* havg   = meta_l + (size_t)B_ * NCHUNK_;        // B*H

  k_havg<<<(B_ * H_) / 256, 256, 0, stream>>>(dec, havg);
  k_scores_partial<<<B_ * NCHUNK_, 256, 0, stream>>>(enc, havg, acc_p, meta_m,
                                                     meta_l);
  k_combine<<<B_, 256, 0, stream>>>(acc_p, meta_m, meta_l, out);
}